// ClassAttention_1580547974514
// MI455X (gfx1250) — compile-verified
//
#include <hip/hip_runtime.h>

#define BATCH 64
#define SEQ   1024
#define CH    1024
#define NH    16
#define HD    64
#define SCALEF 0.125f   // 64^-0.5

typedef __attribute__((ext_vector_type(2))) float v2f;
typedef __attribute__((ext_vector_type(8))) float v8f;

// ---------------------------------------------------------------------------
// CDNA5 async global->LDS helpers (GLOBAL_LOAD_ASYNC_TO_LDS_*, ASYNCcnt).
// LDS address: shared-aperture base has zero low 32 bits (ISA §10.2 aperture
// check uses ADDR[63:32] only; the LDS address is the low 32 bits), so truncating a flat
// pointer to 32 bits gives the workgroup-relative LDS byte address.
// ---------------------------------------------------------------------------
__device__ __forceinline__ unsigned lds_addr(const void* p) {
  return (unsigned)(unsigned long long)p;
}
__device__ __forceinline__ void async_ld_b128(unsigned lds, const void* gptr) {
  asm volatile("global_load_async_to_lds_b128 %0, %1, off"
               :: "v"(lds), "v"(gptr) : "memory");
}
__device__ __forceinline__ void wait_async0() {
  asm volatile("s_wait_asynccnt 0x0" ::: "memory");
}

// ---------------------------------------------------------------------------
// Kernel 1: per batch b
//   q[h,d]   = sum_c x[b,0,c] * W_q[c, h*64+d]          (kept in LDS)
//   u[b,h,c] = SCALE * sum_d W_kv[c, h*64+d] * q[h,d]   (k-half of W_kv)
// ---------------------------------------------------------------------------
__global__ __launch_bounds__(256) void k_qu(const float* __restrict__ x,
                                            const float* __restrict__ Wq,
                                            const float* __restrict__ Wkv,
                                            float* __restrict__ u) {
  __shared__ float xrow[CH];
  __shared__ float qsh[NH * HD];
  const int b = blockIdx.x;
  const int tid = threadIdx.x;
  const float* xb = x + (size_t)b * SEQ * CH;  // token 0 of batch b
  for (int i = tid; i < CH; i += 256) xrow[i] = xb[i];
  __syncthreads();
  for (int j = tid; j < NH * HD; j += 256) {
    float acc = 0.f;
    for (int c = 0; c < CH; ++c) acc = fmaf(xrow[c], Wq[(size_t)c * (NH * HD) + j], acc);
    qsh[j] = acc;
  }
  __syncthreads();
  for (int idx = tid; idx < NH * CH; idx += 256) {
    const int h = idx & (NH - 1);
    const int c = idx >> 4;
    const float* w  = Wkv + (size_t)c * (2 * NH * HD) + h * HD;  // k-columns
    const float* qh = qsh + h * HD;
    float acc = 0.f;
#pragma unroll
    for (int d = 0; d < HD; ++d) acc = fmaf(w[d], qh[d], acc);
    u[((size_t)b * NH + h) * CH + c] = SCALEF * acc;
  }
}

// ---------------------------------------------------------------------------
// Kernel 2: scores[b,h,n] = sum_c x[b,n,c] * u[b,h,c]
// Per-batch GEMM (1024 x 1024) @ (1024 x 16), V_WMMA_F32_16X16X4_F32.
// 8 waves/block, each wave owns a 16-row n-tile; async double-buffered LDS.
// ---------------------------------------------------------------------------
__global__ __launch_bounds__(256) void k_scores(const float* __restrict__ x,
                                                const float* __restrict__ u,
                                                float* __restrict__ scores) {
  __shared__ float xs[2][128 * 36];  // x tile, row stride 36 (banks/align)
  __shared__ float us[2][16 * 36];   // u chunk [h][c]
  const int b   = blockIdx.x >> 3;
  const int rt  = blockIdx.x & 7;
  const int nb  = rt * 128;
  const int tid = threadIdx.x;
  const int w   = tid >> 5;        // wave id (wave32)
  const int l   = tid & 31;        // lane id
  const int l16 = l & 15;
  const int hi2 = (l >= 16) ? 2 : 0;
  const float* xb = x + ((size_t)b * SEQ + nb) * CH;
  const float* ub = u + (size_t)b * NH * CH;

  // async prefetch of one 32-wide K chunk into buffer `buf`
  auto prefetch = [&](int ck, int buf) {
#pragma unroll
    for (int t = 0; t < 4; ++t) {
      const int i = tid + t * 256;          // float4 index, 0..1023
      const int row = i >> 3, c4 = (i & 7) * 4;
      async_ld_b128(lds_addr(&xs[buf][row * 36 + c4]),
                    xb + (size_t)row * CH + ck + c4);
    }
    if (tid < 128) {
      const int h = tid >> 3, c4 = (tid & 7) * 4;
      async_ld_b128(lds_addr(&us[buf][h * 36 + c4]),
                    ub + (size_t)h * CH + ck + c4);
    }
  };

  v8f acc = {};
  prefetch(0, 0);
  for (int ci = 0; ci < CH / 32; ++ci) {
    wait_async0();
    __syncthreads();                 // buffer (ci&1) ready for all waves
    const int cur = ci & 1;
    if (ci + 1 < CH / 32) prefetch((ci + 1) * 32, cur ^ 1);
    const float* xsb = xs[cur];
    const float* usb = us[cur];
#pragma unroll
    for (int kk = 0; kk < 32; kk += 4) {
      // A (16x4, M=n rows, K=c): lanes 0-15 K=0/1, lanes 16-31 K=2/3
      const float* ap = &xsb[(w * 16 + l16) * 36 + kk + hi2];
      // B (4x16, K=c, N=h): lanes 0-15 K=0/1 at col l, lanes 16-31 K=2/3
      const float* bp = &usb[l16 * 36 + kk + hi2];
      v2f a;  a.x = ap[0];  a.y = ap[1];
      v2f bb; bb.x = bp[0]; bb.y = bp[1];
      acc = __builtin_amdgcn_wmma_f32_16x16x4_f32(false, a, false, bb,
                                                  (short)0, acc, false, false);
    }
  }
  // D 16x16: lane -> N(col)=l16 (=h), VGPR r -> M=r (+8 for hi lanes) (=n off)
  const int mofs = (l >= 16) ? 8 : 0;
  float* srow = scores + ((size_t)b * NH + l16) * SEQ + nb + w * 16 + mofs;
#pragma unroll
  for (int r = 0; r < 8; ++r) srow[r] = acc[r];
}

// ---------------------------------------------------------------------------
// Kernel 3: softmax over n (row length 1024), one block per (b,h)
// ---------------------------------------------------------------------------
__global__ __launch_bounds__(256) void k_softmax(float* __restrict__ s) {
  __shared__ float red[256];
  const int tid = threadIdx.x;
  float* p = s + (size_t)blockIdx.x * SEQ;
  float v[4];
  float m = -3.0e38f;
#pragma unroll
  for (int i = 0; i < 4; ++i) { v[i] = p[tid + i * 256]; m = fmaxf(m, v[i]); }
  red[tid] = m; __syncthreads();
  for (int o = 128; o > 0; o >>= 1) {
    if (tid < o) red[tid] = fmaxf(red[tid], red[tid + o]);
    __syncthreads();
  }
  m = red[0]; __syncthreads();
  float sum = 0.f;
#pragma unroll
  for (int i = 0; i < 4; ++i) { v[i] = __expf(v[i] - m); sum += v[i]; }
  red[tid] = sum; __syncthreads();
  for (int o = 128; o > 0; o >>= 1) {
    if (tid < o) red[tid] += red[tid + o];
    __syncthreads();
  }
  const float inv = 1.f / red[0];
#pragma unroll
  for (int i = 0; i < 4; ++i) p[tid + i * 256] = v[i] * inv;
}

// ---------------------------------------------------------------------------
// Kernel 4: y[b,h,c] = sum_n attn[b,h,n] * x[b,n,c]
// Per-batch GEMM (16 x 1024) @ (1024 x 1024), WMMA, async double-buffered.
// ---------------------------------------------------------------------------
__global__ __launch_bounds__(256) void k_attnx(const float* __restrict__ x,
                                               const float* __restrict__ attn,
                                               float* __restrict__ y) {
  __shared__ float xs[2][32 * 144];  // x tile [n row][c], stride 144
  __shared__ float as[2][16 * 36];   // attn chunk [h][n]
  const int b   = blockIdx.x >> 3;
  const int ct  = blockIdx.x & 7;
  const int cb  = ct * 128;
  const int tid = threadIdx.x;
  const int w   = tid >> 5;
  const int l   = tid & 31;
  const int l16 = l & 15;
  const int hi2 = (l >= 16) ? 2 : 0;
  const float* xb = x + (size_t)b * SEQ * CH + cb;
  const float* ab = attn + (size_t)b * NH * SEQ;

  auto prefetch = [&](int nk, int buf) {
#pragma unroll
    for (int t = 0; t < 4; ++t) {
      const int i = tid + t * 256;          // float4 index, 0..1023
      const int row = i >> 5, c4 = (i & 31) * 4;
      async_ld_b128(lds_addr(&xs[buf][row * 144 + c4]),
                    xb + (size_t)(nk + row) * CH + c4);
    }
    if (tid < 128) {
      const int h = tid >> 3, n4 = (tid & 7) * 4;
      async_ld_b128(lds_addr(&as[buf][h * 36 + n4]),
                    ab + (size_t)h * SEQ + nk + n4);
    }
  };

  v8f acc = {};
  prefetch(0, 0);
  for (int ni = 0; ni < SEQ / 32; ++ni) {
    wait_async0();
    __syncthreads();
    const int cur = ni & 1;
    if (ni + 1 < SEQ / 32) prefetch((ni + 1) * 32, cur ^ 1);
    const float* xsb = xs[cur];
    const float* asb = as[cur];
#pragma unroll
    for (int kk = 0; kk < 32; kk += 4) {
      // A (16x4, M=h, K=n)
      const float* ap = &asb[l16 * 36 + kk + hi2];
      v2f a; a.x = ap[0]; a.y = ap[1];
      // B (4x16, K=n, N=c): lanes 0-15 rows kk/kk+1, lanes 16-31 rows kk+2/+3
      v2f bb;
      bb.x = xsb[(kk + hi2    ) * 144 + w * 16 + l16];
      bb.y = xsb[(kk + hi2 + 1) * 144 + w * 16 + l16];
      acc = __builtin_amdgcn_wmma_f32_16x16x4_f32(false, a, false, bb,
                                                  (short)0, acc, false, false);
    }
  }
  const int mofs = (l >= 16) ? 8 : 0;  // h offset
  const int c = cb + w * 16 + l16;
#pragma unroll
  for (int r = 0; r < 8; ++r)
    y[((size_t)b * NH + mofs + r) * CH + c] = acc[r];
}

// ---------------------------------------------------------------------------
// Kernel 5: cls[b,h,d] = sum_c y[b,h,c] * W_kv[c, 1024 + h*64 + d]  (v-half)
//           out[b,:]   = cls[b,:] @ W_proj + b_proj
// Coalesced over W columns: thread tid owns cls outputs idx = tid + 256k,
// whose W_kv column is 1024+idx (consecutive across threads per c row).
// ---------------------------------------------------------------------------
__global__ __launch_bounds__(256) void k_out(const float* __restrict__ y,
                                             const float* __restrict__ Wkv,
                                             const float* __restrict__ Wproj,
                                             const float* __restrict__ bproj,
                                             float* __restrict__ out) {
  __shared__ float ych[NH * 256];   // y chunk [h][c within chunk]
  __shared__ float cls[NH * HD];
  const int b = blockIdx.x;
  const int tid = threadIdx.x;
  float acc[4] = {0.f, 0.f, 0.f, 0.f};
  for (int c0 = 0; c0 < CH; c0 += 256) {
    __syncthreads();
    for (int i = tid; i < NH * 256; i += 256) {
      const int h = i >> 8, c = i & 255;
      ych[h * 256 + c] = y[((size_t)b * NH + h) * CH + c0 + c];
    }
    __syncthreads();
    for (int c = 0; c < 256; ++c) {
      const float* wrow = Wkv + (size_t)(c0 + c) * (2 * NH * HD) + NH * HD;
#pragma unroll
      for (int k = 0; k < 4; ++k) {
        const int idx = tid + 256 * k;           // = h*64 + d
        acc[k] = fmaf(ych[(idx >> 6) * 256 + c], wrow[idx], acc[k]);
      }
    }
  }
#pragma unroll
  for (int k = 0; k < 4; ++k) cls[tid + 256 * k] = acc[k];
  __syncthreads();
  for (int j = tid; j < CH; j += 256) {
    float a2 = bproj[j];
    for (int k = 0; k < NH * HD; ++k)
      a2 = fmaf(cls[k], Wproj[(size_t)k * CH + j], a2);
    out[(size_t)b * CH + j] = a2;
  }
}

// ---------------------------------------------------------------------------
extern "C" void kernel_launch(void* const* d_in, const int* in_sizes, int n_in,
                              void* d_out, int out_size, void* d_ws, size_t ws_size,
                              hipStream_t stream) {
  const float* x     = (const float*)d_in[0];
  const float* Wkv   = (const float*)d_in[1];
  const float* Wq    = (const float*)d_in[2];
  const float* Wproj = (const float*)d_in[3];
  const float* bproj = (const float*)d_in[4];
  float* out = (float*)d_out;

  float* u      = (float*)d_ws;                        // B*NH*CH  = 1M floats
  float* scores = u + (size_t)BATCH * NH * CH;         // B*NH*SEQ = 1M floats
  float* yv     = scores + (size_t)BATCH * NH * SEQ;   // B*NH*CH  = 1M floats

  k_qu     <<<BATCH,      256, 0, stream>>>(x, Wq, Wkv, u);
  k_scores <<<BATCH * 8,  256, 0, stream>>>(x, u, scores);
  k_softmax<<<BATCH * NH, 256, 0, stream>>>(scores);
  k_attnx  <<<BATCH * 8,  256, 0, stream>>>(x, scores, yv);
  k_out    <<<BATCH,      256, 0, stream>>>(yv, Wkv, Wproj, bproj, out);
}